// Aggregator_25056839204917
// MI455X (gfx1250) — compile-verified
//
#include <hip/hip_runtime.h>
#include <cstddef>
#include <cstdint>

// ---------------------------------------------------------------------------
// CDNA5 (gfx1250) implementation of the KG attention aggregator.
// fp32 end-to-end, GEMMs on V_WMMA_F32_16X16X4_F32 (exact fp32 matrix pipe).
// v-GEMM stages its B panel into LDS via GLOBAL_LOAD_ASYNC_TO_LDS_B128.
// ---------------------------------------------------------------------------

typedef __attribute__((ext_vector_type(2))) float v2f;
typedef __attribute__((ext_vector_type(8))) float v8f;

#define D_DIM 128
#define H_DIM 4

__device__ __forceinline__ v8f wmma_f32(v2f a, v2f b, v8f c) {
  // D = A(16x4 f32) * B(4x16 f32) + C(16x16 f32)
  return __builtin_amdgcn_wmma_f32_16x16x4_f32(false, a, false, b, (short)0, c,
                                               false, false);
}

// --------------------------- zero fill -------------------------------------
__global__ void fill_zero(float* __restrict__ p, long long n) {
  long long i = (long long)blockIdx.x * blockDim.x + threadIdx.x;
  long long stride = (long long)gridDim.x * blockDim.x;
  for (; i < n; i += stride) p[i] = 0.0f;
}

// --------------------------- repack B panels -------------------------------
// rp[j*128+n] = {T[2j][n], T[(2j+1)][n]}  -> B fragment becomes ONE b64 load.
__global__ void repack_pairs(const float* __restrict__ qT,
                             const float* __restrict__ kT,
                             const float* __restrict__ vT,
                             v2f* __restrict__ rpq, v2f* __restrict__ rpk,
                             v2f* __restrict__ rpv) {
  const int tid = blockIdx.x * blockDim.x + threadIdx.x;  // 0..8191
  if (tid >= 64 * D_DIM) return;
  const int j = tid >> 7;       // K-pair index 0..63
  const int n = tid & 127;      // column
  const size_t lo = (size_t)(2 * j) * D_DIM + n;
  const size_t hi = (size_t)(2 * j + 1) * D_DIM + n;
  rpq[tid] = (v2f){qT[lo], qT[hi]};
  rpk[tid] = (v2f){kT[lo], kT[hi]};
  rpv[tid] = (v2f){vT[lo], vT[hi]};
}

// --------------------------- pass 1: attention logits ----------------------
// One wave per 16-edge tile. q = emb[head]@qT, k = neigh@kT in WMMA
// C-fragments; per-head q.k dot via in-register lane reduction; writes
// exp_att[e,h] and atomically accumulates att_norm[head,h].
__global__ void __launch_bounds__(128)
kg_attn_logits(const float* __restrict__ ent, const float* __restrict__ wrel,
               const v2f* __restrict__ rpq, const v2f* __restrict__ rpk,
               const int* __restrict__ head, const int* __restrict__ tail,
               const int* __restrict__ etype, float* __restrict__ exp_att,
               float* __restrict__ att_norm, int E, int nTiles) {
  const int wave = threadIdx.x >> 5;
  const int tile = blockIdx.x * (blockDim.x >> 5) + wave;
  if (tile >= nTiles) return;
  const int e0 = tile * 16;
  const int lane = threadIdx.x & 31;
  const int hi = lane >> 4;    // lane-half (selects K pair in A/B frags)
  const int cc = lane & 15;    // N column within tile / M for preloads

  // WGP$/L2 warm-up for the repacked panels (global_prefetch_b8)
  __builtin_prefetch(rpq + (size_t)lane * 256, 0, 3);
  __builtin_prefetch(rpk + (size_t)lane * 256, 0, 3);

  const int eM = min(e0 + cc, E - 1);
  const int hIdx = head[eM];
  const int tIdx = tail[eM];
  const int rIdx = etype[eM] - 1;
  const float* __restrict__ hrow = ent + (size_t)hIdx * D_DIM;
  const float* __restrict__ trow = ent + (size_t)tIdx * D_DIM;
  const float* __restrict__ rrow = wrel + (size_t)rIdx * D_DIM;

  v8f accQ[8];
  v8f accK[8];
#pragma unroll
  for (int nt = 0; nt < 8; ++nt) { accQ[nt] = (v8f)0.0f; accK[nt] = (v8f)0.0f; }

#pragma unroll 1
  for (int kk = 0; kk < D_DIM; kk += 4) {
    const int col = kk + 2 * hi;        // K pair held by this lane-half
    const int j = (kk >> 1) + hi;       // repacked K-pair row
    // A fragments: VGPR0 = K=col, VGPR1 = K=col+1 (16x4 f32 layout)
    v2f aq = *(const v2f*)(hrow + col);
    v2f tv = *(const v2f*)(trow + col);
    v2f rv = *(const v2f*)(rrow + col);
    v2f ak = tv * rv;  // neigh = entity_emb[tail] * rel
#pragma unroll
    for (int nt = 0; nt < 8; ++nt) {
      const int ncol = 16 * nt + cc;
      v2f bq = rpq[(size_t)j * D_DIM + ncol];  // one b64 per B fragment
      v2f bk = rpk[(size_t)j * D_DIM + ncol];
      accQ[nt] = wmma_f32(aq, bq, accQ[nt]);
      accK[nt] = wmma_f32(ak, bk, accK[nt]);
    }
  }

  // att[m,h] = sum_n q[m,n]*k[m,n] over n in [32h,32h+32):
  // element (vgpr r, lane) holds row m = r + 8*hi, col n = 16*nt + cc.
#pragma unroll
  for (int r = 0; r < 8; ++r) {
    const int m = r + 8 * hi;
    const int hm = __shfl(hIdx, m, 32);  // head index of row m
#pragma unroll
    for (int h = 0; h < H_DIM; ++h) {
      float t = accQ[2 * h][r] * accK[2 * h][r] +
                accQ[2 * h + 1][r] * accK[2 * h + 1][r];
      t += __shfl_xor(t, 1, 32);
      t += __shfl_xor(t, 2, 32);
      t += __shfl_xor(t, 4, 32);
      t += __shfl_xor(t, 8, 32);  // reduce over the 16 N-lanes of the group
      float a = fminf(fmaxf(t, -10.0f), 10.0f);
      float ea = __expf(a);
      const int e = e0 + m;
      if (cc == 0 && e < E) {
        exp_att[(size_t)e * H_DIM + h] = ea;
        atomicAdd(&att_norm[(size_t)hm * H_DIM + h], ea);
      }
    }
  }
}

// --------------------------- pass 2: v GEMM + scatter ----------------------
// Stage the repacked vT panel (64KB) into LDS with the gfx1250 async
// global->LDS DMA path, then v = neigh@vT via WMMA with ds_load_b64 B
// fragments; scale by att[m,h] = exp/(norm+1e-8); scatter-add into kg_emb.
__global__ void __launch_bounds__(128)
kg_attn_apply(const float* __restrict__ ent, const float* __restrict__ wrel,
              const v2f* __restrict__ rpv, const int* __restrict__ head,
              const int* __restrict__ tail, const int* __restrict__ etype,
              const float* __restrict__ exp_att,
              const float* __restrict__ att_norm, float* __restrict__ kg_emb,
              int E, int nTiles) {
  __shared__ v2f bpan[64 * D_DIM];  // 8192 * 8B = 64KB repacked vT panel

  // ---- async stage: 128 threads x 32 iters x 16B = 64KB ----
  {
    const uint64_t gbase = (uint64_t)(uintptr_t)rpv;
    const uint32_t lbase = (uint32_t)(uintptr_t)bpan;  // LDS byte offset
#pragma unroll 1
    for (int it = 0; it < 32; ++it) {
      const uint32_t off = (uint32_t)(it * 128 + (int)threadIdx.x) * 16u;
      const uint64_t ga = gbase + off;
      const uint32_t la = lbase + off;
      asm volatile("global_load_async_to_lds_b128 %0, %1, off"
                   :: "v"(la), "v"(ga)
                   : "memory");
    }
    asm volatile("s_wait_asynccnt 0x0" ::: "memory");
  }
  __syncthreads();  // staged panel visible to all waves in the workgroup

  const int wave = threadIdx.x >> 5;
  const int tile = blockIdx.x * (blockDim.x >> 5) + wave;
  if (tile >= nTiles) return;
  const int e0 = tile * 16;
  const int lane = threadIdx.x & 31;
  const int hi = lane >> 4;
  const int cc = lane & 15;

  const int eM = min(e0 + cc, E - 1);
  const int hIdx = head[eM];
  const int tIdx = tail[eM];
  const int rIdx = etype[eM] - 1;
  const float* __restrict__ trow = ent + (size_t)tIdx * D_DIM;
  const float* __restrict__ rrow = wrel + (size_t)rIdx * D_DIM;

  // per-lane (m = cc) normalized attention for all 4 heads
  float attv[H_DIM];
  {
    const float* ea = exp_att + (size_t)eM * H_DIM;
    const float* an = att_norm + (size_t)hIdx * H_DIM;
#pragma unroll
    for (int h = 0; h < H_DIM; ++h) attv[h] = ea[h] / (an[h] + 1e-8f);
  }

  v8f accV[8];
#pragma unroll
  for (int nt = 0; nt < 8; ++nt) accV[nt] = (v8f)0.0f;

#pragma unroll 1
  for (int kk = 0; kk < D_DIM; kk += 4) {
    const int col = kk + 2 * hi;
    const int j = (kk >> 1) + hi;
    v2f tv = *(const v2f*)(trow + col);
    v2f rv = *(const v2f*)(rrow + col);
    v2f av = tv * rv;  // neigh fragment
#pragma unroll
    for (int nt = 0; nt < 8; ++nt) {
      const int ncol = 16 * nt + cc;
      v2f bv = bpan[(size_t)j * D_DIM + ncol];  // ds_load_b64 from LDS
      accV[nt] = wmma_f32(av, bv, accV[nt]);
    }
  }

  // element (nt, r, lane): row m = r + 8*hi, col n = 16*nt + cc, head h=nt>>1
#pragma unroll
  for (int nt = 0; nt < 8; ++nt) {
    const int h = nt >> 1;  // 32 columns per attention head
#pragma unroll
    for (int r = 0; r < 8; ++r) {
      const int m = r + 8 * hi;
      const float sc = __shfl(attv[h], m, 32);
      const int hm = __shfl(hIdx, m, 32);
      const int e = e0 + m;
      const float val = accV[nt][r] * sc;
      if (e < E)
        atomicAdd(&kg_emb[(size_t)hm * D_DIM + 16 * nt + cc], val);
    }
  }
}

// --------------------------- pass 3: edge weight + segment max -------------
// w = ||kg[head]*rel||^2 * ||kg[tail]*rel||^2  ( == (hr*tr)^2 )
// One wave per edge; atomicMax on bit pattern (valid: w >= 0).
__global__ void edge_w_max(const float* __restrict__ kg,
                           const float* __restrict__ wrel,
                           const int* __restrict__ head,
                           const int* __restrict__ tail,
                           const int* __restrict__ etype,
                           float* __restrict__ w_arr, float* __restrict__ m_arr,
                           int E) {
  const int wid = (int)(((long long)blockIdx.x * blockDim.x + threadIdx.x) >> 5);
  if (wid >= E) return;
  const int lane = threadIdx.x & 31;
  const int h = head[wid];
  const int t = tail[wid];
  const int r = etype[wid] - 1;
  const float4 kh = *(const float4*)(kg + (size_t)h * D_DIM + lane * 4);
  const float4 kt = *(const float4*)(kg + (size_t)t * D_DIM + lane * 4);
  const float4 rv = *(const float4*)(wrel + (size_t)r * D_DIM + lane * 4);
  float hx = kh.x * rv.x, hy = kh.y * rv.y, hz = kh.z * rv.z, hw = kh.w * rv.w;
  float tx = kt.x * rv.x, ty = kt.y * rv.y, tz = kt.z * rv.z, tw = kt.w * rv.w;
  float hp = hx * hx + hy * hy + hz * hz + hw * hw;
  float tp = tx * tx + ty * ty + tz * tz + tw * tw;
#pragma unroll
  for (int m = 1; m < 32; m <<= 1) {
    hp += __shfl_xor(hp, m, 32);
    tp += __shfl_xor(tp, m, 32);
  }
  if (lane == 0) {
    const float w = hp * tp;  // (hr*tr)^2 with hr=sqrt(hp), tr=sqrt(tp)
    w_arr[wid] = w;
    atomicMax((unsigned int*)&m_arr[h], __float_as_uint(w));
  }
}

// --------------------------- pass 4: exp + segment sum ---------------------
__global__ void edge_exp_sum(const float* __restrict__ w_arr,
                             const float* __restrict__ m_arr,
                             const int* __restrict__ head,
                             float* __restrict__ es, float* __restrict__ s_arr,
                             int E) {
  const int e = blockIdx.x * blockDim.x + threadIdx.x;
  if (e >= E) return;
  const int h = head[e];
  const float ev = __expf(w_arr[e] - m_arr[h]);
  es[e] = ev;
  atomicAdd(&s_arr[h], ev);
}

// --------------------------- pass 5: final weighted scatter ----------------
__global__ void edge_scatter_out(const float* __restrict__ ent,
                                 const int* __restrict__ head,
                                 const int* __restrict__ tail,
                                 const float* __restrict__ es,
                                 const float* __restrict__ s_arr,
                                 float* __restrict__ out, int E) {
  const int wid = (int)(((long long)blockIdx.x * blockDim.x + threadIdx.x) >> 5);
  if (wid >= E) return;
  const int lane = threadIdx.x & 31;
  const int h = head[wid];
  const int t = tail[wid];
  const float soft = es[wid] / s_arr[h];
  const float4 tv = *(const float4*)(ent + (size_t)t * D_DIM + lane * 4);
  float* o = out + (size_t)h * D_DIM + lane * 4;
  atomicAdd(o + 0, soft * tv.x);
  atomicAdd(o + 1, soft * tv.y);
  atomicAdd(o + 2, soft * tv.z);
  atomicAdd(o + 3, soft * tv.w);
}

// ---------------------------------------------------------------------------
extern "C" void kernel_launch(void* const* d_in, const int* in_sizes, int n_in,
                              void* d_out, int out_size, void* d_ws,
                              size_t ws_size, hipStream_t stream) {
  (void)n_in; (void)out_size; (void)ws_size;
  const float* ent  = (const float*)d_in[0];   // entity_emb [N,128]
  const float* wrel = (const float*)d_in[3];   // weight     [32,128]
  const float* qT   = (const float*)d_in[4];   // qTrans     [128,128]
  const float* kT   = (const float*)d_in[5];   // kTrans     [128,128]
  const float* vT   = (const float*)d_in[6];   // vTrans     [128,128]
  const int* eidx   = (const int*)d_in[7];     // edge_index [2,E]
  const int* etype  = (const int*)d_in[8];     // edge_type  [E]
  float* out = (float*)d_out;

  const int E = in_sizes[8];
  const int NENT = in_sizes[0] / D_DIM;
  const int* head = eidx;
  const int* tail = eidx + E;

  // workspace layout (floats)
  float* ws = (float*)d_ws;
  size_t off = 0;
  float* exp_att  = ws + off; off += (size_t)E * H_DIM;       // [E,4]
  float* att_norm = ws + off; off += (size_t)NENT * H_DIM;    // [N,4]
  float* kg_emb   = ws + off; off += (size_t)NENT * D_DIM;    // [N,128]
  float* w_arr    = ws + off; off += (size_t)E;               // [E]
  float* m_arr    = ws + off; off += (size_t)NENT;            // [N]
  float* s_arr    = ws + off; off += (size_t)NENT;            // [N]
  float* es       = ws + off; off += (size_t)E;               // [E]
  v2f* rpq = (v2f*)(ws + off); off += (size_t)64 * D_DIM * 2; // paired qT
  v2f* rpk = (v2f*)(ws + off); off += (size_t)64 * D_DIM * 2; // paired kT
  v2f* rpv = (v2f*)(ws + off); off += (size_t)64 * D_DIM * 2; // paired vT

  // zero accumulators (att_norm+kg_emb contiguous; m+s contiguous) and out
  fill_zero<<<2048, 256, 0, stream>>>(att_norm,
                                      (long long)NENT * (H_DIM + D_DIM));
  fill_zero<<<512, 256, 0, stream>>>(m_arr, (long long)NENT * 2);
  fill_zero<<<2048, 256, 0, stream>>>(out, (long long)NENT * D_DIM);

  // repack transform matrices into K-pair panels (one b64 per B fragment)
  repack_pairs<<<(64 * D_DIM + 255) / 256, 256, 0, stream>>>(qT, kT, vT, rpq,
                                                             rpk, rpv);

  const int nTiles = (E + 15) / 16;
  const int wavesPerBlk = 4;  // 128 threads
  const int gemmBlocks = (nTiles + wavesPerBlk - 1) / wavesPerBlk;

  kg_attn_logits<<<gemmBlocks, 128, 0, stream>>>(
      ent, wrel, rpq, rpk, head, tail, etype, exp_att, att_norm, E, nTiles);

  kg_attn_apply<<<gemmBlocks, 128, 0, stream>>>(
      ent, wrel, rpv, head, tail, etype, exp_att, att_norm, kg_emb, E, nTiles);

  const int wavesBlocks = (E + 7) / 8;  // 8 waves / 256-thread block
  edge_w_max<<<wavesBlocks, 256, 0, stream>>>(kg_emb, wrel, head, tail, etype,
                                              w_arr, m_arr, E);

  edge_exp_sum<<<(E + 255) / 256, 256, 0, stream>>>(w_arr, m_arr, head, es,
                                                    s_arr, E);

  edge_scatter_out<<<wavesBlocks, 256, 0, stream>>>(ent, head, tail, es, s_arr,
                                                    out, E);
}